// FuzzyLayer_53687091200221
// MI455X (gfx1250) — compile-verified
//
#include <hip/hip_runtime.h>

// FuzzyLayer for MI455X (gfx1250, wave32).
// Bandwidth-bound pipeline (~0.5 GiB min traffic -> ~23us @ 23.3 TB/s).
//  K1 conv1: one wave per row, float4 column strips + packed f32 FMA (vmem-lean).
//  K2 BN1+fuzzify, K2b shift-correlation stats, K3 analytic BN2 prep.
//  K4 conv2 as fp32 WMMA implicit GEMM + fused BN2 (output written exactly once),
//     fz tile staged into LDS via GLOBAL_LOAD_ASYNC_TO_LDS when available.

#define HH 256
#define WW 256
#define NN 16
#define CCH 64
#define NHW 1048576.0f  // 16*256*256 positions per channel

typedef __attribute__((ext_vector_type(2))) float v2f;
typedef __attribute__((ext_vector_type(8))) float v8f;

#if __has_builtin(__builtin_amdgcn_global_load_async_to_lds_b32)
#define FUZZY_ASYNC_LDS 1
typedef __attribute__((address_space(1))) int gint;  // clang prints as '__device__ int'
typedef __attribute__((address_space(3))) int lint;  // clang prints as '__shared__ int'
#endif

__device__ __forceinline__ float4 f4z() { float4 z; z.x = z.y = z.z = z.w = 0.f; return z; }
__device__ __forceinline__ float4 f4fma(float4 a, float s, float4 c) {
  c.x = fmaf(a.x, s, c.x); c.y = fmaf(a.y, s, c.y);
  c.z = fmaf(a.z, s, c.z); c.w = fmaf(a.w, s, c.w);
  return c;
}

// ---------------------------------------------------------------------------
// K1: conv1 (64ch -> 1ch, 3x3 SAME) + bias -> h_raw, plus BN1 sum/sumsq.
// One wave per output row; lane owns 8 columns (two float4 strips).
// Per channel: 6x global_load_b128, q_dw column partials, 2 shuffles for halo.
// ---------------------------------------------------------------------------
__global__ __launch_bounds__(256) void k1_conv1_stats(
    const float* __restrict__ x, const float* __restrict__ w1,
    const float* __restrict__ b1, float* __restrict__ hraw,
    float* __restrict__ stats /* [0]=sum [1]=sumsq */) {
  const int wave = threadIdx.x >> 5;
  const int lane = threadIdx.x & 31;
  const int R = blockIdx.x * 8 + wave;  // global row 0..4095
  const int n = R >> 8;
  const int r = R & 255;
  const bool rm = r > 0, rp = r < HH - 1;
  const float* base = x + ((size_t)n * CCH * HH + (size_t)r) * WW + lane * 8;

  float4 acc0 = f4z(), acc1 = f4z();
  for (int c = 0; c < CCH; ++c) {
    const float* rowp = base + (size_t)c * (HH * WW);
    const float4* p0 = (const float4*)rowp;
    float4 x00 = p0[0], x01 = p0[1];
    float4 xm0 = f4z(), xm1 = f4z(), xp0 = f4z(), xp1 = f4z();
    if (rm) { const float4* p = (const float4*)(rowp - WW); xm0 = p[0]; xm1 = p[1]; }
    if (rp) { const float4* p = (const float4*)(rowp + WW); xp0 = p[0]; xp1 = p[1]; }

    const float* wk = w1 + c * 9;  // uniform -> scalar loads
    const float w0 = wk[0], w1c = wk[1], w2c = wk[2];
    const float w3 = wk[3], w4 = wk[4], w5 = wk[5];
    const float w6 = wk[6], w7 = wk[7], w8 = wk[8];

    // q_dw(col) = sum_dh wk[dh*3+dw] * x_dh(col)
    float4 q00 = f4fma(xm0, w0, f4fma(x00, w3, f4fma(xp0, w6, f4z())));
    float4 q01 = f4fma(xm1, w0, f4fma(x01, w3, f4fma(xp1, w6, f4z())));
    float4 q10 = f4fma(xm0, w1c, f4fma(x00, w4, f4fma(xp0, w7, f4z())));
    float4 q11 = f4fma(xm1, w1c, f4fma(x01, w4, f4fma(xp1, w7, f4z())));
    float4 q20 = f4fma(xm0, w2c, f4fma(x00, w5, f4fma(xp0, w8, f4z())));
    float4 q21 = f4fma(xm1, w2c, f4fma(x01, w5, f4fma(xp1, w8, f4z())));

    // halo across lanes: left = prev lane's q0 at its col 7; right = next lane's q2 at col 0
    float left = __shfl_up(q01.w, 1, 32);
    if (lane == 0) left = 0.f;
    float right = __shfl_down(q20.x, 1, 32);
    if (lane == 31) right = 0.f;

    // y(c) = q0(c-1) + q1(c) + q2(c+1)
    acc0.x += q10.x + left   + q20.y;
    acc0.y += q10.y + q00.x  + q20.z;
    acc0.z += q10.z + q00.y  + q20.w;
    acc0.w += q10.w + q00.z  + q21.x;
    acc1.x += q11.x + q00.w  + q21.y;
    acc1.y += q11.y + q01.x  + q21.z;
    acc1.z += q11.z + q01.y  + q21.w;
    acc1.w += q11.w + q01.z  + right;
  }
  const float bb = b1[0];
  acc0.x += bb; acc0.y += bb; acc0.z += bb; acc0.w += bb;
  acc1.x += bb; acc1.y += bb; acc1.z += bb; acc1.w += bb;

  float4* hp = (float4*)(hraw + (size_t)R * WW + lane * 8);
  hp[0] = acc0; hp[1] = acc1;

  float s = acc0.x + acc0.y + acc0.z + acc0.w + acc1.x + acc1.y + acc1.z + acc1.w;
  float s2 = acc0.x * acc0.x + acc0.y * acc0.y + acc0.z * acc0.z + acc0.w * acc0.w +
             acc1.x * acc1.x + acc1.y * acc1.y + acc1.z * acc1.z + acc1.w * acc1.w;
  #pragma unroll
  for (int off = 16; off > 0; off >>= 1) {
    s += __shfl_xor(s, off, 32);
    s2 += __shfl_xor(s2, off, 32);
  }
  __shared__ float ls, ls2;
  if (threadIdx.x == 0) { ls = 0.f; ls2 = 0.f; }
  __syncthreads();
  if (lane == 0) { atomicAdd(&ls, s); atomicAdd(&ls2, s2); }
  __syncthreads();
  if (threadIdx.x == 0) { atomicAdd(&stats[0], ls); atomicAdd(&stats[1], ls2); }
}

// ---------------------------------------------------------------------------
// K2: apply BN1 (batch stats) then the 4-set sequential fuzzify. Writes fz.
// ---------------------------------------------------------------------------
__global__ __launch_bounds__(256) void k2_bn1_fuzzify(
    const float* __restrict__ hraw, const float* __restrict__ stats,
    const float* __restrict__ g1, const float* __restrict__ be1,
    const float* __restrict__ lb, const float* __restrict__ ub,
    float* __restrict__ fz) {
  const size_t i = (size_t)blockIdx.x * 256 + threadIdx.x;
  const float inv = 1.f / NHW;
  const float mean = stats[0] * inv;
  const float var = stats[1] * inv - mean * mean;
  const float is = rsqrtf(var + 1e-5f);
  float xv = (hraw[i] - mean) * is * g1[0] + be1[0];
  float fuzz = 0.f;
  #pragma unroll
  for (int f = 0; f < 4; ++f) {
    const float l = lb[f], u = ub[f];
    const float ld = xv - l, ud = u - xv;
    const bool inr = (ld > 0.f) && (ud > 0.f);
    const bool outr = (ld < 0.f) || (ud < 0.f);
    const float xr = 2.f * l - xv;
    const float e_ld = __expf(-0.5f * ld * ld);
    const float c_in = e_ld + __expf(-0.5f * ud * ud);
    const float t = u - xr;
    const float c_out = e_ld + __expf(-0.5f * t * t);
    fuzz += inr ? c_in : (outr ? c_out : 0.f);
    xv = outr ? xr : xv;
  }
  fz[i] = fuzz;
}

// ---------------------------------------------------------------------------
// K2b: shift-correlation stats over fz: S[k]=sum a_k, M[k<=k']=sum a_k*a_k'
// (a_k = zero-padded 3x3 shifts). Channel-independent -> 54 scalars.
// ---------------------------------------------------------------------------
__global__ __launch_bounds__(256) void k2b_fz_stats(
    const float* __restrict__ fz, float* __restrict__ S, float* __restrict__ M) {
  float s[9] = {};
  float m[45] = {};
  const int tid = threadIdx.x;
  for (int it = 0; it < 16; ++it) {
    const int idx = (blockIdx.x * 16 + it) * 256 + tid;  // 0 .. 1M-1
    const int w = idx & 255;
    const int r = (idx >> 8) & 255;
    const int n = idx >> 16;
    const float* p = fz + ((size_t)n << 16);
    float a[9];
    #pragma unroll
    for (int k = 0; k < 9; ++k) {
      const int rr = r + k / 3 - 1, cc = w + k % 3 - 1;
      a[k] = (rr >= 0 && rr < HH && cc >= 0 && cc < WW) ? p[rr * WW + cc] : 0.f;
    }
    int mi = 0;
    #pragma unroll
    for (int k = 0; k < 9; ++k) {
      s[k] += a[k];
      #pragma unroll
      for (int k2 = k; k2 < 9; ++k2) m[mi++] += a[k] * a[k2];
    }
  }
  #pragma unroll
  for (int k = 0; k < 9; ++k)
    for (int off = 16; off > 0; off >>= 1) s[k] += __shfl_xor(s[k], off, 32);
  for (int k = 0; k < 45; ++k)
    for (int off = 16; off > 0; off >>= 1) m[k] += __shfl_xor(m[k], off, 32);
  __shared__ float red[54];
  if (tid < 54) red[tid] = 0.f;
  __syncthreads();
  if ((tid & 31) == 0) {
    for (int k = 0; k < 9; ++k) atomicAdd(&red[k], s[k]);
    for (int k = 0; k < 45; ++k) atomicAdd(&red[9 + k], m[k]);
  }
  __syncthreads();
  if (tid < 9) atomicAdd(&S[tid], red[tid]);
  else if (tid < 54) atomicAdd(&M[tid - 9], red[tid]);
}

// ---------------------------------------------------------------------------
// K3: per-channel BN2 scale/shift from analytic stats.
// y = b2 + t, t = sum_k w_k a_k  =>  var_c = E[t^2]-E[t]^2.
// ---------------------------------------------------------------------------
__global__ void k3_bn2_prep(
    const float* __restrict__ S, const float* __restrict__ Mu,
    const float* __restrict__ w2, const float* __restrict__ b2,
    const float* __restrict__ g2, const float* __restrict__ be2,
    float* __restrict__ scale, float* __restrict__ shift) {
  const int c = threadIdx.x;
  if (c >= CCH) return;
  const float inv = 1.f / NHW;
  float wv[9];
  #pragma unroll
  for (int k = 0; k < 9; ++k) wv[k] = w2[c * 9 + k];
  float Et = 0.f;
  #pragma unroll
  for (int k = 0; k < 9; ++k) Et += wv[k] * S[k];
  Et *= inv;
  float Et2 = 0.f;
  #pragma unroll
  for (int k = 0; k < 9; ++k)
    #pragma unroll
    for (int k2 = k; k2 < 9; ++k2) {
      const int idx = k * 9 + k2 - k * (k + 1) / 2;  // upper-tri storage
      Et2 += ((k == k2) ? 1.f : 2.f) * wv[k] * wv[k2] * (Mu[idx] * inv);
    }
  const float var = Et2 - Et * Et;
  const float is = rsqrtf(var + 1e-5f);
  const float sc = g2[c] * is;
  scale[c] = sc;
  shift[c] = be2[c] - Et * sc;  // mean_c = b2_c + Et; acc excludes b2 -> cancels
}

// ---------------------------------------------------------------------------
// K4: conv2 (1->64ch) as fp32 WMMA implicit GEMM, fused BN2, writes d_out once.
// D(16ch x 16cols) = A(16ch x 4k) * B(4k x 16cols), K=9 padded to 12 (3 blocks),
// 4 M-tiles over 64 channels. Wave = one output row x 16 cols; block = 8 rows.
// fz tile staged into LDS via async load-to-LDS; OOB halo lanes pull from a
// known-zero scratch word (no LDS pre-zero -> no DS/ASYNC ordering hazard).
// ---------------------------------------------------------------------------
__global__ __launch_bounds__(256) void k4_conv2_bn_wmma(
    const float* __restrict__ fz, const float* __restrict__ w2,
    const float* __restrict__ scale, const float* __restrict__ shift,
    const float* __restrict__ zerow, float* __restrict__ out) {
  const int bid = blockIdx.x;
  const int ct = bid & 15;         // 16 col tiles of 16
  const int rt = (bid >> 4) & 31;  // 32 row tiles of 8
  const int n = bid >> 9;          // 16 images
  const int c0 = ct * 16, r0 = rt * 8;

  __shared__ float tile[10 * 18];  // fz rows r0-1..r0+8, cols c0-1..c0+16
  __shared__ float sS[CCH], sB[CCH];
  const int tid = threadIdx.x;
  if (tid < CCH) { sS[tid] = scale[tid]; sB[tid] = shift[tid]; }
  const float* fzp = fz + ((size_t)n << 16);
  if (tid < 180) {
    const int tr = tid / 18, tc = tid % 18;
    const int gr = r0 - 1 + tr, gc = c0 - 1 + tc;
    const bool ok = (gr >= 0 && gr < HH && gc >= 0 && gc < WW);
    const float* src = ok ? (fzp + gr * WW + gc) : zerow;
#ifdef FUZZY_ASYNC_LDS
    __builtin_amdgcn_global_load_async_to_lds_b32((gint*)src, (lint*)&tile[tid], 0, 0);
#else
    tile[tid] = *src;
#endif
  }
#ifdef FUZZY_ASYNC_LDS
#if __has_builtin(__builtin_amdgcn_s_wait_asynccnt)
  __builtin_amdgcn_s_wait_asynccnt(0);
#else
  asm volatile("s_wait_asynccnt 0" ::: "memory");
#endif
#endif
  __syncthreads();

  const int wave = tid >> 5;  // wave -> output row r0+wave
  const int lane = tid & 31;
  const int lh = lane >> 4;   // lane half: K pair select / M high half
  const int ll = lane & 15;

  // A fragments (weights), M=channel: 16x4 fp32 layout
  // VGPR0 holds K={4kb, 4kb+2} across lane halves; VGPR1 holds +1.
  v2f A[4][3];
  #pragma unroll
  for (int mt = 0; mt < 4; ++mt)
    #pragma unroll
    for (int kb = 0; kb < 3; ++kb) {
      const int ch = mt * 16 + ll;
      const int k0 = kb * 4 + lh * 2;
      v2f av;
      av[0] = (k0 < 9) ? w2[ch * 9 + k0] : 0.f;
      av[1] = (k0 + 1 < 9) ? w2[ch * 9 + k0 + 1] : 0.f;
      A[mt][kb] = av;
    }

  // B fragments (fz patches) from LDS, 4x16 fp32 (mirror of A)
  v2f B[3];
  #pragma unroll
  for (int kb = 0; kb < 3; ++kb) {
    const int k0 = kb * 4 + lh * 2;
    v2f bv;
    bv[0] = 0.f;
    bv[1] = 0.f;
    if (k0 < 9) bv[0] = tile[(wave + k0 / 3) * 18 + ll + k0 % 3];
    if (k0 + 1 < 9) {
      const int k = k0 + 1;
      bv[1] = tile[(wave + k / 3) * 18 + ll + k % 3];
    }
    B[kb] = bv;
  }

  float* outp = out + (size_t)n * CCH * HH * WW + (size_t)(r0 + wave) * WW + c0;
  #pragma unroll
  for (int mt = 0; mt < 4; ++mt) {
    v8f acc = {0.f, 0.f, 0.f, 0.f, 0.f, 0.f, 0.f, 0.f};
    #pragma unroll
    for (int kb = 0; kb < 3; ++kb)
      acc = __builtin_amdgcn_wmma_f32_16x16x4_f32(
          false, A[mt][kb], false, B[kb], (short)0, acc, false, false);
    #pragma unroll
    for (int v = 0; v < 8; ++v) {
      const int ch = mt * 16 + v + lh * 8;  // D layout: M = v + 8*laneHalf
      outp[(size_t)ch * (HH * WW) + ll] = acc[v] * sS[ch] + sB[ch];
    }
  }
}

// ---------------------------------------------------------------------------
extern "C" void kernel_launch(void* const* d_in, const int* in_sizes, int n_in,
                              void* d_out, int out_size, void* d_ws, size_t ws_size,
                              hipStream_t stream) {
  const float* x   = (const float*)d_in[0];
  const float* w1  = (const float*)d_in[1];
  const float* b1  = (const float*)d_in[2];
  const float* w2  = (const float*)d_in[3];
  const float* b2  = (const float*)d_in[4];
  const float* g1  = (const float*)d_in[5];
  const float* be1 = (const float*)d_in[6];
  const float* g2  = (const float*)d_in[7];
  const float* be2 = (const float*)d_in[8];
  const float* lb  = (const float*)d_in[9];
  const float* ub  = (const float*)d_in[10];
  float* out = (float*)d_out;

  float* ws    = (float*)d_ws;
  float* stats = ws;        // [0..1] BN1 sum/sumsq
  float* S     = ws + 8;    // 9 shift sums
  float* Mu    = ws + 20;   // 45 upper-tri shift correlations
  float* scale = ws + 80;   // 64
  float* shift = ws + 160;  // 64
  float* zerow = ws + 240;  // stays 0 (inside memset region, never written)
  float* hraw  = (float*)((char*)d_ws + 4096);               // 4 MiB
  float* fz    = (float*)((char*)d_ws + 4096 + (1u << 22));  // 4 MiB

  (void)hipMemsetAsync(d_ws, 0, 1024, stream);  // zero accumulators (capture-safe)
  k1_conv1_stats<<<512, 256, 0, stream>>>(x, w1, b1, hraw, stats);
  k2_bn1_fuzzify<<<4096, 256, 0, stream>>>(hraw, stats, g1, be1, lb, ub, fz);
  k2b_fz_stats<<<256, 256, 0, stream>>>(fz, S, Mu);
  k3_bn2_prep<<<1, 64, 0, stream>>>(S, Mu, w2, b2, g2, be2, scale, shift);
  k4_conv2_bn_wmma<<<NN * 32 * 16, 256, 0, stream>>>(fz, w2, scale, shift, zerow, out);
}